// EmbeddingLayer_20641612825101
// MI455X (gfx1250) — compile-verified
//
#include <hip/hip_runtime.h>
#include <math.h>

typedef __attribute__((ext_vector_type(2))) float v2f;
typedef __attribute__((ext_vector_type(8))) float v8f;

#define N_NODES 2000
#define F_IN    128
#define HEADS   8
#define GDIM    64
#define DMODEL  512
#define BATCH   64
#define SEQ_S   512
#define SEQ_T   512
#define LEAKY   0.2f
#define KCHUNK  200   // attn K-chunk staged in LDS (200*64*4 = 50 KB)
#define TWO_PI_OVER_DAY (6.283185307179586f / 86400.0f)

// workspace layout (float offsets)
#define WS_H    0         // [H][N][G]   = 1,024,000 f
#define WS_ES   1024000   // [H][N]      = 16,000 f
#define WS_ED   1040000
#define WS_RMAX 1056000
#define WS_RSUM 1072000
#define WS_GEMB 1088000   // [N][D]      = 1,024,000 f

// Async-stage `bytes` (multiple of 2048) from contiguous global `src` into LDS
// starting at 32-bit LDS offset `ldsbase`; all 128 threads of the block help.
// Uses GLOBAL_LOAD_ASYNC_TO_LDS_B128 (ASYNCcnt) + s_wait_asynccnt.
__device__ __forceinline__ void async_stage(unsigned ldsbase, const char* src,
                                            int n2k, int tid) {
  for (int i = 0; i < n2k; ++i) {
    unsigned off = (unsigned)(i * 2048 + tid * 16);
    asm volatile("global_load_async_to_lds_b128 %0, %1, off"
                 :: "v"(ldsbase + off), "v"(src + off) : "memory");
  }
  asm volatile("s_wait_asynccnt 0x0" ::: "memory");
}

// ---------------------------------------------------------------------------
// K1: h[head,n,g] = x[n,:] @ W_gat[head,:,g]  as 2000x512 = 2000x128 @ 128x512
// V_WMMA_F32_16X16X4_F32, one wave per 16x16 C tile.
// ---------------------------------------------------------------------------
__global__ void k_gat_h(const float* __restrict__ x, const float* __restrict__ Wg,
                        float* __restrict__ h) {
  const int w    = blockIdx.x * 4 + (threadIdx.x >> 5);   // 4000 waves
  const int lane = threadIdx.x & 31;
  const int mt   = w >> 5;                                // 0..124
  const int nt   = w & 31;                                // 0..31
  const int row0 = mt * 16;
  const int col0 = nt * 16;
  const int koff = (lane < 16) ? 0 : 2;                   // A/B K sub-offset
  const int c    = col0 + (lane & 15);                    // flat output column
  const int head = c >> 6;
  const int g    = c & 63;
  const float* xrow = x + (size_t)(row0 + (lane & 15)) * F_IN;
  const float* wcol = Wg + head * (F_IN * GDIM) + g;      // stride GDIM per f

  v8f acc = {};
  for (int kb = 0; kb < F_IN; kb += 4) {
    v2f a, b;
    a.x = xrow[kb + koff];
    a.y = xrow[kb + koff + 1];
    b.x = wcol[(size_t)(kb + koff) * GDIM];
    b.y = wcol[(size_t)(kb + koff + 1) * GDIM];
    acc = __builtin_amdgcn_wmma_f32_16x16x4_f32(false, a, false, b,
                                                (short)0, acc, false, false);
  }
  const int mbase = (lane < 16) ? 0 : 8;
#pragma unroll
  for (int r = 0; r < 8; ++r) {
    int n = row0 + mbase + r;
    h[head * (N_NODES * GDIM) + (size_t)n * GDIM + g] = acc[r];
  }
}

// ---------------------------------------------------------------------------
// K2: es[h,n] = h[h,n,:]·a_src[h,:],  ed likewise
// ---------------------------------------------------------------------------
__global__ void k_es_ed(const float* __restrict__ h, const float* __restrict__ asrc,
                        const float* __restrict__ adst,
                        float* __restrict__ es, float* __restrict__ ed) {
  int i = blockIdx.x * blockDim.x + threadIdx.x;        // 16000 items
  if (i >= HEADS * N_NODES) return;
  int head = i / N_NODES, n = i % N_NODES;
  const float* hp = h + head * (N_NODES * GDIM) + (size_t)n * GDIM;
  const float* as = asrc + head * GDIM;
  const float* ad = adst + head * GDIM;
  float s = 0.f, d = 0.f;
  for (int g = 0; g < GDIM; ++g) { float v = hp[g]; s += v * as[g]; d += v * ad[g]; }
  es[i] = s; ed[i] = d;
}

// ---------------------------------------------------------------------------
// K3: per (head,row) masked online softmax stats (max, sum) — flash style,
// one wave per row, wave32 xor-shuffle reduction.
// ---------------------------------------------------------------------------
__global__ void k_rowstats(const float* __restrict__ adj, const float* __restrict__ es,
                           const float* __restrict__ ed,
                           float* __restrict__ rmax, float* __restrict__ rsum) {
  const int w    = blockIdx.x * 4 + (threadIdx.x >> 5);   // 16000 waves
  const int lane = threadIdx.x & 31;
  const int head = w / N_NODES, i = w % N_NODES;
  const float esi  = es[head * N_NODES + i];
  const float* arow = adj + (size_t)i * N_NODES;
  const float* edh  = ed + head * N_NODES;
  float m = -3.0e38f, s = 0.f;
  for (int j = lane; j < N_NODES; j += 32) {
    if (arow[j] > 0.f) {
      float t = esi + edh[j];
      float l = t > 0.f ? t : LEAKY * t;
      if (l > m) { s *= __expf(m - l); m = l; }
      s += __expf(l - m);
    }
  }
  for (int off = 16; off > 0; off >>= 1) {
    float m2 = __shfl_xor(m, off, 32);
    float s2 = __shfl_xor(s, off, 32);
    float mn = fmaxf(m, m2);
    s = s * __expf(m - mn) + s2 * __expf(m2 - mn);
    m = mn;
  }
  if (lane == 0) { rmax[w] = m; rsum[w] = s; }
}

// ---------------------------------------------------------------------------
// K4: gemb[n, head*64+g] = relu( softmax(e) @ h ).  Attention weights are
// regenerated inside the WMMA K-loop (never materialized).  The 4 waves of a
// block share (head, 16-row tile); the B operand (h chunk, contiguous) is
// async-staged into LDS in 50 KB slices and consumed via ds_load.
// 8*125 blocks x 4 waves (one per 16-col group), K=2000 in chunks of 200.
// ---------------------------------------------------------------------------
__global__ void k_attn_gemm(const float* __restrict__ adj, const float* __restrict__ h,
                            const float* __restrict__ es, const float* __restrict__ ed,
                            const float* __restrict__ rmax, const float* __restrict__ rsum,
                            float* __restrict__ gemb) {
  __shared__ float tileH[KCHUNK * GDIM];                  // 50 KB
  const int w    = blockIdx.x * 4 + (threadIdx.x >> 5);
  const int lane = threadIdx.x & 31;
  const int cg   = w & 3;                                 // differs per wave
  const int rt   = (w >> 2) % 125;                        // shared in block
  const int head = w / (125 * 4);                         // shared in block
  const int row0 = rt * 16;
  const int g0   = cg * 16;
  const int koff = (lane < 16) ? 0 : 2;
  const int myrow = row0 + (lane & 15);
  const float esr  = es[head * N_NODES + myrow];
  const float rmx  = rmax[head * N_NODES + myrow];
  const float* edh  = ed + head * N_NODES;
  const float* hh   = h + head * (N_NODES * GDIM);
  const float* arow = adj + (size_t)myrow * N_NODES;
  const int gcol = g0 + (lane & 15);
  const unsigned ldsbase = (unsigned)(unsigned long long)(void*)tileH;

  v8f acc = {};
  for (int j0 = 0; j0 < N_NODES; j0 += KCHUNK) {
    // stage h[j0 .. j0+199][0..63] (contiguous 51200 B) into LDS
    async_stage(ldsbase, (const char*)(hh + (size_t)j0 * GDIM),
                (KCHUNK * GDIM * 4) / 2048, threadIdx.x);
    __syncthreads();
#pragma unroll 2
    for (int j = 0; j < KCHUNK; j += 4) {
      const int k0 = j0 + j + koff;
      float ad0 = arow[k0], ad1 = arow[k0 + 1];
      float e0 = esr + edh[k0], e1 = esr + edh[k0 + 1];
      e0 = e0 > 0.f ? e0 : LEAKY * e0;
      e1 = e1 > 0.f ? e1 : LEAKY * e1;
      v2f a, b;
      a.x = ad0 > 0.f ? __expf(e0 - rmx) : 0.f;   // unnormalized attention
      a.y = ad1 > 0.f ? __expf(e1 - rmx) : 0.f;
      b.x = tileH[(j + koff) * GDIM + gcol];
      b.y = tileH[(j + koff + 1) * GDIM + gcol];
      acc = __builtin_amdgcn_wmma_f32_16x16x4_f32(false, a, false, b,
                                                  (short)0, acc, false, false);
    }
    __syncthreads();                               // before next chunk overwrite
  }
  const int mbase = (lane < 16) ? 0 : 8;
#pragma unroll
  for (int r = 0; r < 8; ++r) {
    int n = row0 + mbase + r;
    float v = acc[r] / rsum[head * N_NODES + n];   // softmax normalization
    v = v > 0.f ? v : 0.f;                          // final relu
    gemb[(size_t)n * DMODEL + head * GDIM + g0 + (lane & 15)] = v;
  }
}

// ---------------------------------------------------------------------------
// K5: trip half: out = relu(trip @ W_trip) + [sin,cos](depart)*mask @ W_time + b
// 32768x512 = 32768x128 @ 128x512 WMMA GEMM.  The 4 waves of a block share the
// 16-row A tile (contiguous 8 KB) -> async-staged into LDS once per block.
// ---------------------------------------------------------------------------
__global__ void k_trip(const float* __restrict__ trip, const float* __restrict__ Wt,
                       const float* __restrict__ tmask, const float* __restrict__ depart,
                       const float* __restrict__ Wtime, const float* __restrict__ btime,
                       float* __restrict__ out) {
  __shared__ float tileA[16 * F_IN];                      // 8 KB
  const int bb   = blockIdx.x;                            // 16384 blocks
  const int warp = threadIdx.x >> 5;
  const int lane = threadIdx.x & 31;
  const int mt   = bb >> 3;                               // shared by 4 waves
  const int nt   = ((bb & 7) << 2) + warp;                // 0..31
  const int row0 = mt * 16, col0 = nt * 16;
  const unsigned ldsbase = (unsigned)(unsigned long long)(void*)tileA;

  // rows row0..row0+15 are contiguous (row-major [32768,128])
  async_stage(ldsbase, (const char*)(trip + (size_t)row0 * F_IN), 4, threadIdx.x);
  __syncthreads();

  const int koff = (lane < 16) ? 0 : 2;
  const int d = col0 + (lane & 15);
  const float* arow = &tileA[(lane & 15) * F_IN];

  v8f acc = {};
  for (int kb = 0; kb < F_IN; kb += 4) {
    __builtin_prefetch(Wt + (size_t)(kb + 4) * DMODEL + d, 0, 1);
    v2f a, b;
    a.x = arow[kb + koff];
    a.y = arow[kb + koff + 1];
    b.x = Wt[(size_t)(kb + koff) * DMODEL + d];
    b.y = Wt[(size_t)(kb + koff + 1) * DMODEL + d];
    acc = __builtin_amdgcn_wmma_f32_16x16x4_f32(false, a, false, b,
                                                (short)0, acc, false, false);
  }
  const float w0 = Wtime[d], w1 = Wtime[DMODEL + d], bt = btime[d];
  const int mbase = (lane < 16) ? 0 : 8;
#pragma unroll
  for (int r = 0; r < 8; ++r) {
    int row = row0 + mbase + r;                 // flat (b*T + t)
    float ang = depart[row] * TWO_PI_OVER_DAY;
    float mk  = tmask[row];
    float sv = __sinf(ang) * mk, cv = __cosf(ang) * mk;
    float v = acc[r]; v = v > 0.f ? v : 0.f;
    out[(size_t)row * DMODEL + d] = v + sv * w0 + cv * w1 + bt;
  }
}

// ---------------------------------------------------------------------------
// K6: region half: gather graph_emb rows by (index-1), add cyclical time + bias
// ---------------------------------------------------------------------------
__global__ void k_region(const float* __restrict__ gemb, const int* __restrict__ idx,
                         const float* __restrict__ rmask, const float* __restrict__ arrive,
                         const float* __restrict__ Wtime, const float* __restrict__ btime,
                         float* __restrict__ out) {
  const int p   = blockIdx.x;                   // 0..B*S-1
  const int tid = threadIdx.x;                  // 256 threads, 2 d's each
  const int ia  = idx[p];
  const float mk  = rmask[p];
  const float ang = arrive[p] * TWO_PI_OVER_DAY;
  const float sv = __sinf(ang) * mk, cv = __cosf(ang) * mk;
  const float* grow = gemb + (size_t)((ia > 0) ? (ia - 1) : 0) * DMODEL;
#pragma unroll
  for (int q = 0; q < 2; ++q) {
    int d = tid + q * 256;
    float g = (ia > 0) ? grow[d] : 0.f;
    out[(size_t)p * DMODEL + d] = g + sv * Wtime[d] + cv * Wtime[DMODEL + d] + btime[d];
  }
}

// ---------------------------------------------------------------------------
extern "C" void kernel_launch(void* const* d_in, const int* in_sizes, int n_in,
                              void* d_out, int out_size, void* d_ws, size_t ws_size,
                              hipStream_t stream) {
  const float* region_batch = (const float*)d_in[0];
  const float* trip_batch   = (const float*)d_in[1];
  const float* trip_mask    = (const float*)d_in[2];
  const float* region_mask  = (const float*)d_in[3];
  const float* graph_mask   = (const float*)d_in[4];
  const float* arrive       = (const float*)d_in[5];
  const float* depart       = (const float*)d_in[6];
  const int*   index_array  = (const int*)d_in[7];
  const float* W_trip       = (const float*)d_in[8];
  const float* W_gat        = (const float*)d_in[9];
  const float* a_src        = (const float*)d_in[10];
  const float* a_dst        = (const float*)d_in[11];
  const float* W_time       = (const float*)d_in[12];
  const float* b_time       = (const float*)d_in[13];

  float* ws    = (float*)d_ws;
  float* h     = ws + WS_H;
  float* es    = ws + WS_ES;
  float* ed    = ws + WS_ED;
  float* rmax  = ws + WS_RMAX;
  float* rsum  = ws + WS_RSUM;
  float* gemb  = ws + WS_GEMB;

  float* out_region = (float*)d_out;
  float* out_trip   = (float*)d_out + (size_t)BATCH * SEQ_S * DMODEL;

  // GAT pipeline
  k_gat_h    <<<1000, 128, 0, stream>>>(region_batch, W_gat, h);
  k_es_ed    <<<125,  128, 0, stream>>>(h, a_src, a_dst, es, ed);
  k_rowstats <<<4000, 128, 0, stream>>>(graph_mask, es, ed, rmax, rsum);
  k_attn_gemm<<<1000, 128, 0, stream>>>(graph_mask, h, es, ed, rmax, rsum, gemb);
  k_region   <<<BATCH * SEQ_S, 256, 0, stream>>>(gemb, index_array, region_mask,
                                                 arrive, W_time, b_time, out_region);
  // Independent trip half
  k_trip     <<<16384, 128, 0, stream>>>(trip_batch, W_trip, trip_mask, depart,
                                         W_time, b_time, out_trip);
}